// XGBackpropLayer_76398878261907
// MI455X (gfx1250) — compile-verified
//
#include <hip/hip_runtime.h>
#include <hip/hip_bf16.h>

// Problem constants (from reference)
#define B_TOTAL 16384
#define F       256
#define T       100
#define DEPTH   6
#define N_INT   63
#define N_LEAF  64

// Tiling
#define TILE_B          64      // samples per block
#define THREADS         256     // 8 waves (wave32)
#define TREE_GROUPS     4
#define TREES_PER_GROUP 25      // 4 * 25 = 100 trees
#define ROW_STRIDE      257     // 256 floats + 1 pad dword (inserted by TDM pad logic)

// LDS layout (dynamic shared, base offset 0)
#define ROWS_OFF    0
#define ROWS_BYTES  (TILE_B * ROW_STRIDE * 4)      // 65792
#define NODES_OFF   ROWS_BYTES                     // 65792 (8B aligned)
#define NODES_BYTES (T * N_INT * 8)                // 50400
#define LEAF_OFF    (NODES_OFF + NODES_BYTES)      // 116192
#define LEAF_BYTES  (T * N_LEAF * 4)               // 25600
#define PART_OFF    (LEAF_OFF + LEAF_BYTES)        // 141792
#define SMEM_BYTES  (PART_OFF + THREADS * 4)       // 142816 -> 2 blocks / 320KB WGP

typedef unsigned int u32x4 __attribute__((ext_vector_type(4)));
typedef int          i32x8 __attribute__((ext_vector_type(8)));
typedef int          i32x4 __attribute__((ext_vector_type(4)));

__global__ void __launch_bounds__(THREADS)
xgb_forest_kernel(const float* __restrict__ input,
                  const float* __restrict__ split_cond,
                  const float* __restrict__ leaf_gain,
                  const float* __restrict__ base_score,
                  const int*   __restrict__ split_idx,
                  float* __restrict__ out)
{
    extern __shared__ char smem[];
    float* rows   = (float*)(smem + ROWS_OFF);   // [TILE_B][ROW_STRIDE]
    int2*  nodes  = (int2*) (smem + NODES_OFF);  // [T][N_INT] {feat_idx, cond_bits}
    float* leaves = (float*)(smem + LEAF_OFF);   // [T][N_LEAF]
    float* part   = (float*)(smem + PART_OFF);   // [TREE_GROUPS][TILE_B]

    const int tid = threadIdx.x;

    // ---------------------------------------------------------------
    // TDM: async-copy this block's 64x256 f32 tile (64 KB) into LDS.
    // pad_enable with pad_interval=7 (256 dwords) and pad_amount=0 (1 dword)
    // gives an LDS row stride of 257 dwords -> bank = (sample + feat) % 64.
    // TDM ignores EXEC, so issue from exactly one wave (wave-uniform branch).
    // ---------------------------------------------------------------
    if (tid < 32) {
        unsigned long long gaddr =
            (unsigned long long)(const void*)(input + (size_t)blockIdx.x * TILE_B * F);

        u32x4 g0;
        g0[0] = 1u;                                        // count=1 (valid), user mode
        g0[1] = 0u;                                        // lds_addr = 0 (rows at LDS base)
        g0[2] = (unsigned)(gaddr & 0xFFFFFFFFull);         // global_addr[31:0]
        g0[3] = (unsigned)((gaddr >> 32) & 0x01FFFFFFull)  // global_addr[56:32]
              | (2u << 30);                                // type = 2 ("image")

        i32x8 g1;
        g1[0] = (2 << 16)      // data_size = 4 bytes
              | (1 << 20)      // pad_enable
              | (7 << 22)      // pad_interval: 256 dwords (one 256-float row)
              | (0 << 25);     // pad_amount: 1 dword
        g1[1] = (int)((unsigned)F << 16);       // tensor_dim0[15:0] = 256 (bits 63:48)
        g1[2] = (int)((unsigned)TILE_B << 16);  // tensor_dim1[15:0] = 64  (bits 95:80)
        g1[3] = (int)((unsigned)F << 16);       // tile_dim0 = 256 (bits 127:112)
        g1[4] = TILE_B;                         // tile_dim1 = 64; tile_dim2 = 0 (2D)
        g1[5] = F;                              // tensor_dim0_stride = 256 (lo 32 of 48)
        g1[6] = 0;                              // stride hi bits / dim1_stride lo = 0
        g1[7] = 0;

        i32x4 gz4 = {0, 0, 0, 0};               // groups 2/3 unused for 2D tile
        i32x8 gz8 = {0, 0, 0, 0, 0, 0, 0, 0};   // extra group (clang-23 6-arg form)
        __builtin_amdgcn_tensor_load_to_lds(g0, g1, gz4, gz4, gz8, 0);
    }

    // ---------------------------------------------------------------
    // Stage tree tables into LDS (overlaps the TDM transfer).
    // Interleave {feat_idx, cond} so traversal does one ds_load_b64 per level.
    // ---------------------------------------------------------------
    for (int i = tid; i < T * N_INT; i += THREADS) {
        int2 v;
        v.x = split_idx[i];
        v.y = __float_as_int(split_cond[i]);
        nodes[i] = v;
    }
    for (int i = tid; i < T * N_LEAF; i += THREADS) {
        leaves[i] = leaf_gain[i];
    }

    if (tid < 32) {
        __builtin_amdgcn_s_wait_tensorcnt(0);   // wave 0: TDM tile landed in LDS
    }
    __syncthreads();                            // everyone: tile + tables visible

    // ---------------------------------------------------------------
    // Traversal: thread = (sample s_local, tree group g); 25 trees each.
    // node -> 2*node + 1 + (feat > cond); leaf = node - 63 after 6 levels.
    // ---------------------------------------------------------------
    const int s_local = tid & (TILE_B - 1);
    const int g       = tid >> 6;
    const float* __restrict__ row = rows + s_local * ROW_STRIDE;
    const int t0 = g * TREES_PER_GROUP;

    float acc = 0.0f;
    for (int t = t0; t < t0 + TREES_PER_GROUP; ++t) {
        const int2* __restrict__ tn = nodes + t * N_INT;
        int node = 0;
#pragma unroll
        for (int d = 0; d < DEPTH; ++d) {
            int2  nc = tn[node];                // ds_load_b64: {feat_idx, cond}
            float f  = row[nc.x];               // ds_load_b32 gather (padded stride)
            node = 2 * node + 1 + ((f > __int_as_float(nc.y)) ? 1 : 0);
        }
        acc += leaves[t * N_LEAF + (node - N_INT)];
    }

    part[tid] = acc;                            // part[g * TILE_B + s_local]
    __syncthreads();

    if (tid < TILE_B) {
        float r = part[tid] + part[tid + TILE_B] + part[tid + 2 * TILE_B]
                + part[tid + 3 * TILE_B] + base_score[0];
        out[(size_t)blockIdx.x * TILE_B + tid] = r;
    }
}

extern "C" void kernel_launch(void* const* d_in, const int* in_sizes, int n_in,
                              void* d_out, int out_size, void* d_ws, size_t ws_size,
                              hipStream_t stream)
{
    (void)in_sizes; (void)n_in; (void)out_size; (void)d_ws; (void)ws_size;

    const float* input      = (const float*)d_in[0];
    const float* split_cond = (const float*)d_in[1];
    const float* leaf_gain  = (const float*)d_in[2];
    const float* base_score = (const float*)d_in[3];
    const int*   split_idx  = (const int*)  d_in[4];
    float*       out        = (float*)d_out;

    // > 64 KB dynamic LDS: raise the cap (deterministic, host-side, not captured)
    (void)hipFuncSetAttribute((const void*)xgb_forest_kernel,
                              hipFuncAttributeMaxDynamicSharedMemorySize,
                              SMEM_BYTES);

    dim3 grid(B_TOTAL / TILE_B);   // 256 blocks
    dim3 block(THREADS);           // 256 threads = 8 waves
    xgb_forest_kernel<<<grid, block, SMEM_BYTES, stream>>>(
        input, split_cond, leaf_gain, base_score, split_idx, out);
}